// TripletSeAttention_87024627351662
// MI455X (gfx1250) — compile-verified
//
#include <hip/hip_runtime.h>
#include <math.h>

typedef float v2f __attribute__((ext_vector_type(2)));
typedef float v8f __attribute__((ext_vector_type(8)));

#define BB 32
#define HH 56
#define WW 56
#define CC 256
#define CH 32
#define BN_EPS 1e-3f
#define SLAB (WW * CC)          // 14336 floats per (b,h) slab
#define PITCH 257               // LDS pitch to kill bank conflicts

// ---------------------------------------------------------------------------
// Kernel 1: per (b,h) slab, pool over W (-> pmax1/pavg1[b,h,c]) and over C
// (-> pmax3/pavg3[b,h,w]). One read of x, staged through LDS.
// ---------------------------------------------------------------------------
__global__ __launch_bounds__(256) void pool_w_c_kernel(
    const float* __restrict__ x,
    float* __restrict__ pmax1, float* __restrict__ pavg1,
    float* __restrict__ pmax3, float* __restrict__ pavg3) {
  __shared__ float tile[WW * PITCH];
  const int bh = blockIdx.x;
  const int t = threadIdx.x;
  const float* xp = x + (size_t)bh * SLAB;

  // Coalesced B128 global loads -> padded LDS
  for (int i4 = t; i4 < SLAB / 4; i4 += 256) {
    float4 v = ((const float4*)xp)[i4];
    int idx = i4 * 4;
    int w = idx >> 8, c = idx & 255;
    float* d = &tile[w * PITCH + c];
    d[0] = v.x; d[1] = v.y; d[2] = v.z; d[3] = v.w;
  }
  __syncthreads();

  // pool over W (t = c)
  {
    float mx = -INFINITY, sm = 0.f;
    for (int w = 0; w < WW; ++w) {
      float v = tile[w * PITCH + t];
      mx = fmaxf(mx, v);
      sm += v;
    }
    pmax1[(size_t)bh * CC + t] = mx;
    pavg1[(size_t)bh * CC + t] = sm * (1.0f / WW);
  }
  // pool over C (t = w, first 56 threads)
  if (t < WW) {
    float mx = -INFINITY, sm = 0.f;
    for (int c = 0; c < CC; ++c) {
      float v = tile[t * PITCH + c];
      mx = fmaxf(mx, v);
      sm += v;
    }
    pmax3[(size_t)bh * WW + t] = mx;
    pavg3[(size_t)bh * WW + t] = sm * (1.0f / CC);
  }
}

// ---------------------------------------------------------------------------
// Kernel 2: pool over H -> pmax2/pavg2[b,c,w]. One block per (b,w).
// Second pass over x: hits the 192MB L2.
// ---------------------------------------------------------------------------
__global__ __launch_bounds__(256) void pool_h_kernel(
    const float* __restrict__ x,
    float* __restrict__ pmax2, float* __restrict__ pavg2) {
  const int b = blockIdx.x / WW;
  const int w = blockIdx.x % WW;
  const int c = threadIdx.x;
  const float* p = x + ((size_t)b * HH * WW + w) * CC + c;
  float mx = -INFINITY, sm = 0.f;
  for (int h = 0; h < HH; ++h) {
    if (h + 1 < HH) __builtin_prefetch(p + (size_t)(h + 1) * WW * CC, 0, 1);
    float v = p[(size_t)h * WW * CC];
    mx = fmaxf(mx, v);
    sm += v;
  }
  pmax2[((size_t)b * CC + c) * WW + w] = mx;
  pavg2[((size_t)b * CC + c) * WW + w] = sm * (1.0f / HH);
}

// ---------------------------------------------------------------------------
// Kernel 3: generic 7x7 2ch->1ch SAME conv + BN + sigmoid over (D1,D2) maps.
// 16x16 output tile per block, 22x22x2 halo staged in LDS.
// Output written with arbitrary strides (branch 2 writes transposed a2t).
// ---------------------------------------------------------------------------
__global__ __launch_bounds__(256) void conv_att_kernel(
    const float* __restrict__ pmax, const float* __restrict__ pavg,
    int D1, int D2,
    const float* __restrict__ wt, const float* __restrict__ bias,
    const float* __restrict__ bng, const float* __restrict__ bnb,
    const float* __restrict__ bnm, const float* __restrict__ bnv,
    float* __restrict__ out, int os1, int os2, int oplane) {
  __shared__ float smx[22 * 22];
  __shared__ float sav[22 * 22];
  const int b = blockIdx.z;
  const int t1 = blockIdx.y * 16;
  const int t2 = blockIdx.x * 16;
  const int tx = threadIdx.x & 15;
  const int ty = threadIdx.x >> 4;
  const size_t ibase = (size_t)b * D1 * D2;

  for (int i = threadIdx.x; i < 22 * 22; i += 256) {
    int r = i / 22, cc = i % 22;
    int d1 = t1 + r - 3, d2 = t2 + cc - 3;
    bool ok = (d1 >= 0) && (d1 < D1) && (d2 >= 0) && (d2 < D2);
    size_t idx = ibase + (size_t)d1 * D2 + d2;
    smx[i] = ok ? pmax[idx] : 0.f;
    sav[i] = ok ? pavg[idx] : 0.f;
  }
  __syncthreads();

  const int d1 = t1 + ty, d2 = t2 + tx;
  if (d1 < D1 && d2 < D2) {
    float acc = 0.f;
#pragma unroll
    for (int kh = 0; kh < 7; ++kh)
#pragma unroll
      for (int kw = 0; kw < 7; ++kw) {
        int si = (ty + kh) * 22 + (tx + kw);
        int wi = (kh * 7 + kw) * 2;
        acc = fmaf(smx[si], wt[wi + 0], acc);
        acc = fmaf(sav[si], wt[wi + 1], acc);
      }
    float rs = rsqrtf(bnv[0] + BN_EPS);
    float y = (acc + bias[0] - bnm[0]) * rs * bng[0] + bnb[0];
    out[(size_t)b * oplane + (size_t)d1 * os1 + (size_t)d2 * os2] =
        1.0f / (1.0f + expf(-y));
  }
}

// ---------------------------------------------------------------------------
// Kernel 4a: partial GAP of s = x*(a1+a2+a3) over w for each (b,h,c).
// ---------------------------------------------------------------------------
__global__ __launch_bounds__(256) void gap_partial_kernel(
    const float* __restrict__ x, const float* __restrict__ a1,
    const float* __restrict__ a2t, const float* __restrict__ a3,
    float* __restrict__ part) {
  const int bh = blockIdx.x;
  const int b = bh / HH;
  const int c = threadIdx.x;
  const float* xp = x + (size_t)bh * SLAB;
  const float a1v = a1[(size_t)bh * CC + c];
  float sm = 0.f;
  for (int w = 0; w < WW; ++w) {
    float a3v = a3[(size_t)bh * WW + w];                        // uniform
    float a2v = a2t[((size_t)b * WW + w) * CC + c];             // coalesced
    sm += xp[(size_t)w * CC + c] * (a1v + a2v + a3v);
  }
  part[(size_t)bh * CC + c] = sm;
}

// Kernel 4b: reduce partials over h -> gap[b,c] (deterministic, no atomics)
__global__ __launch_bounds__(256) void gap_reduce_kernel(
    const float* __restrict__ part, float* __restrict__ gap) {
  const int b = blockIdx.x;
  const int c = threadIdx.x;
  float s = 0.f;
  for (int h = 0; h < HH; ++h) s += part[((size_t)b * HH + h) * CC + c];
  gap[(size_t)b * CC + c] = s * (1.0f / (HH * WW));
}

// ---------------------------------------------------------------------------
// Kernel 5: SE via V_WMMA_F32_16X16X4_F32.
//  GEMM1: h = relu(gap(32x256) @ W1(256x32) + b1)   -> LDS
//  GEMM2: cw = sigmoid(h(32x32) @ W2(32x256) + b2)
// Layouts per CDNA5 ISA 7.12.2 (wave32):
//  A 16x4: lane L -> M = L&15, VGPR v -> K = v + 2*(L>>4)
//  B 4x16 / C,D 16x16: VGPR v, group g=L>>4 -> row = v + (nrows/2)*g, col = L&15
// ---------------------------------------------------------------------------
__global__ __launch_bounds__(256) void se_wmma_kernel(
    const float* __restrict__ gap,
    const float* __restrict__ w1, const float* __restrict__ b1,
    const float* __restrict__ w2, const float* __restrict__ b2,
    float* __restrict__ cw) {
  __shared__ float hbuf[32 * 32];
  const int lane = threadIdx.x & 31;
  const int wave = threadIdx.x >> 5;
  const int g = lane >> 4;
  const int ln = lane & 15;

  // ---- GEMM1: 2x2 tiles of 16x16, K=256 (64 steps of 4). Waves 0..3.
  if (wave < 4) {
    const int tm = wave >> 1, tn = wave & 1;
    v8f acc = {};
    for (int kb = 0; kb < 64; ++kb) {
      const int k0 = kb * 4 + 2 * g;
      v2f a, bm;
      const int m = tm * 16 + ln;
      a.x = gap[(size_t)m * CC + k0 + 0];
      a.y = gap[(size_t)m * CC + k0 + 1];
      const int n = tn * 16 + ln;
      bm.x = w1[(size_t)(k0 + 0) * CH + n];
      bm.y = w1[(size_t)(k0 + 1) * CH + n];
      acc = __builtin_amdgcn_wmma_f32_16x16x4_f32(
          false, a, false, bm, (short)0, acc, false, false);
    }
#pragma unroll
    for (int v = 0; v < 8; ++v) {
      const int m = tm * 16 + v + 8 * g;
      const int n = tn * 16 + ln;
      float hv = acc[v] + b1[n];
      hbuf[m * 32 + n] = hv > 0.f ? hv : 0.f;
    }
  }
  __syncthreads();

  // ---- GEMM2: 2x16 tiles of 16x16, K=32 (8 steps). 8 waves x 4 tiles.
  for (int tidx = wave; tidx < 32; tidx += 8) {
    const int tm = tidx >> 4;   // 0..1
    const int tn = tidx & 15;   // 0..15
    v8f acc = {};
    for (int kb = 0; kb < 8; ++kb) {
      const int k0 = kb * 4 + 2 * g;
      v2f a, bm;
      const int m = tm * 16 + ln;
      a.x = hbuf[m * 32 + k0 + 0];
      a.y = hbuf[m * 32 + k0 + 1];
      const int n = tn * 16 + ln;
      bm.x = w2[(size_t)(k0 + 0) * CC + n];
      bm.y = w2[(size_t)(k0 + 1) * CC + n];
      acc = __builtin_amdgcn_wmma_f32_16x16x4_f32(
          false, a, false, bm, (short)0, acc, false, false);
    }
#pragma unroll
    for (int v = 0; v < 8; ++v) {
      const int m = tm * 16 + v + 8 * g;
      const int n = tn * 16 + ln;
      float yv = acc[v] + b2[n];
      cw[(size_t)m * CC + n] = 1.0f / (1.0f + expf(-yv));
    }
  }
}

// ---------------------------------------------------------------------------
// Kernel 6: out = x*(a1+a2+a3)*cw, vectorized float4 (B128 load/store).
// Thread t: c-quad (t&63)*4, w-phase t>>6 (4 phases of 14 w's).
// ---------------------------------------------------------------------------
__global__ __launch_bounds__(256) void final_kernel(
    const float* __restrict__ x, const float* __restrict__ a1,
    const float* __restrict__ a2t, const float* __restrict__ a3,
    const float* __restrict__ cw, float* __restrict__ out) {
  const int bh = blockIdx.x;
  const int b = bh / HH;
  const int t = threadIdx.x;
  const int c4 = (t & 63) << 2;
  const int wq = t >> 6;
  const float* xp = x + (size_t)bh * SLAB;
  float* op = out + (size_t)bh * SLAB;
  const float4 a1v = *(const float4*)(a1 + (size_t)bh * CC + c4);
  const float4 cwv = *(const float4*)(cw + (size_t)b * CC + c4);
  for (int w = wq; w < WW; w += 4) {
    const float a3v = a3[(size_t)bh * WW + w];
    const float4 a2v = *(const float4*)(a2t + ((size_t)b * WW + w) * CC + c4);
    const float4 xv = *(const float4*)(xp + (size_t)w * CC + c4);
    float4 o;
    o.x = xv.x * (a1v.x + a2v.x + a3v) * cwv.x;
    o.y = xv.y * (a1v.y + a2v.y + a3v) * cwv.y;
    o.z = xv.z * (a1v.z + a2v.z + a3v) * cwv.z;
    o.w = xv.w * (a1v.w + a2v.w + a3v) * cwv.w;
    *(float4*)(op + (size_t)w * CC + c4) = o;
  }
}

// ---------------------------------------------------------------------------
extern "C" void kernel_launch(void* const* d_in, const int* in_sizes, int n_in,
                              void* d_out, int out_size, void* d_ws,
                              size_t ws_size, hipStream_t stream) {
  const float* x = (const float*)d_in[0];
  // conv/bn param sets: base index 1 + 6*(i-1): w, b, g, beta, mean, var
  const float* c1w = (const float*)d_in[1];
  const float* c1b = (const float*)d_in[2];
  const float* g1 = (const float*)d_in[3];
  const float* be1 = (const float*)d_in[4];
  const float* m1 = (const float*)d_in[5];
  const float* v1 = (const float*)d_in[6];
  const float* c2w = (const float*)d_in[7];
  const float* c2b = (const float*)d_in[8];
  const float* g2 = (const float*)d_in[9];
  const float* be2 = (const float*)d_in[10];
  const float* m2 = (const float*)d_in[11];
  const float* v2 = (const float*)d_in[12];
  const float* c3w = (const float*)d_in[13];
  const float* c3b = (const float*)d_in[14];
  const float* g3 = (const float*)d_in[15];
  const float* be3 = (const float*)d_in[16];
  const float* m3 = (const float*)d_in[17];
  const float* v3 = (const float*)d_in[18];
  const float* sew1 = (const float*)d_in[19];
  const float* seb1 = (const float*)d_in[20];
  const float* sew2 = (const float*)d_in[21];
  const float* seb2 = (const float*)d_in[22];
  float* out = (float*)d_out;

  // workspace carve (floats)
  const size_t NHC = (size_t)BB * HH * CC;  // 458752
  const size_t NCW = (size_t)BB * CC * WW;  // 458752
  const size_t NHW = (size_t)BB * HH * WW;  // 100352
  float* ws = (float*)d_ws;
  float* pmax1 = ws;              ws += NHC;
  float* pavg1 = ws;              ws += NHC;
  float* pmax2 = ws;              ws += NCW;
  float* pavg2 = ws;              ws += NCW;
  float* pmax3 = ws;              ws += NHW;
  float* pavg3 = ws;              ws += NHW;
  float* a1 = ws;                 ws += NHC;
  float* a2t = ws;                ws += NCW;  // [b][w][c] (transposed)
  float* a3 = ws;                 ws += NHW;
  float* part = ws;               ws += NHC;
  float* gap = ws;                ws += (size_t)BB * CC;
  float* cw = ws;                 ws += (size_t)BB * CC;

  const int BHBLK = BB * HH;  // 1792

  // 1) pools over W and C (single x read)
  pool_w_c_kernel<<<BHBLK, 256, 0, stream>>>(x, pmax1, pavg1, pmax3, pavg3);
  // 2) pool over H (x now L2-resident)
  pool_h_kernel<<<BB * WW, 256, 0, stream>>>(x, pmax2, pavg2);

  // 3) three attention convs
  // branch 1: (D1=H, D2=C) -> a1[b,h,c]
  conv_att_kernel<<<dim3(CC / 16, HH / 16 + 1, BB), 256, 0, stream>>>(
      pmax1, pavg1, HH, CC, c1w, c1b, g1, be1, m1, v1, a1, CC, 1, HH * CC);
  // branch 2: (D1=C, D2=W) -> a2t[b,w,c] (transposed write)
  conv_att_kernel<<<dim3(WW / 16 + 1, CC / 16, BB), 256, 0, stream>>>(
      pmax2, pavg2, CC, WW, c2w, c2b, g2, be2, m2, v2, a2t, 1, CC, WW * CC);
  // branch 3: (D1=H, D2=W) -> a3[b,h,w]
  conv_att_kernel<<<dim3(WW / 16 + 1, HH / 16 + 1, BB), 256, 0, stream>>>(
      pmax3, pavg3, HH, WW, c3w, c3b, g3, be3, m3, v3, a3, WW, 1, HH * WW);

  // 4) GAP of s (two-stage, deterministic)
  gap_partial_kernel<<<BHBLK, 256, 0, stream>>>(x, a1, a2t, a3, part);
  gap_reduce_kernel<<<BB, 256, 0, stream>>>(part, gap);

  // 5) SE GEMMs on WMMA
  se_wmma_kernel<<<1, 256, 0, stream>>>(gap, sew1, seb1, sew2, seb2, cw);

  // 6) final fused multiply (only compulsory HBM write)
  final_kernel<<<BHBLK, 256, 0, stream>>>(x, a1, a2t, a3, cw, out);
}